// FasterRCNN_69123203662609
// MI455X (gfx1250) — compile-verified
//
#include <hip/hip_runtime.h>
#include <math.h>

// Problem constants (match reference)
#define BATCH   2
#define NPROP   1000
#define NCLS    81
#define KPAD    84                   // K padded to 21 chunks of 4
#define CCLS    80
#define ROWS    (BATCH * NPROP)      // 2000
#define SCORE_T 0.05f
#define NMS_T   0.5f
#define DETCAP  100

typedef __attribute__((ext_vector_type(2))) float v2f;
typedef __attribute__((ext_vector_type(8))) float v8f;
typedef __attribute__((ext_vector_type(4))) int   v4i;

// Address-space-qualified pointer types for the async-to-LDS builtins:
// signature (from clang diagnostics): (__device__ intN*, __local__ intN*, imm, imm)
typedef __attribute__((address_space(1))) int   as1_i32;
typedef __attribute__((address_space(3))) int   as3_i32;
typedef __attribute__((address_space(1))) v4i   as1_i128;
typedef __attribute__((address_space(3))) v4i   as3_i128;

// ---- CDNA5 feature guards (compile-safe on host pass & older toolchains) ----
#if defined(__gfx1250__) && __has_builtin(__builtin_amdgcn_wmma_f32_16x16x4_f32)
#define USE_WMMA 1
#else
#define USE_WMMA 0
#endif

#if defined(__gfx1250__) && \
    __has_builtin(__builtin_amdgcn_global_load_async_to_lds_b32) && \
    __has_builtin(__builtin_amdgcn_global_load_async_to_lds_b128) && \
    __has_builtin(__builtin_amdgcn_s_wait_asynccnt)
#define USE_ASYNC 1
#else
#define USE_ASYNC 0
#endif

// =====================================================================
// Kernel 1: softmax scores (drop background) + box decode, written to
// workspace in [B, C, N] / [B, C, N, 4] (class-major) layout.
// One wave owns a 16-row tile.  exp(logit - rowmax) is staged once into
// an LDS tile (unconditional clamped loads; masking on the stored value
// so no divergent save-exec around loads), then the softmax denominator
// is computed on the matrix pipe: V_WMMA_F32_16X16X4_F32 with an all-ones
// B matrix, accumulating 21 K-chunks; A fragments come straight from LDS.
// =====================================================================
__global__ __launch_bounds__(256)
void k_scores_decode(const float* __restrict__ logits,   // [ROWS, 81]
                     const float* __restrict__ breg,     // [ROWS, 320]
                     const float* __restrict__ prop,     // [ROWS, 4]
                     float* __restrict__ scores_ws,      // [B*C, N]
                     float* __restrict__ boxes_ws)       // [B*C, N, 4]
{
    __shared__ float exl[8][16][KPAD];  // exp(logit - m), zero-padded (42 KB)
    __shared__ float tile[8][16][16];   // per-wave 16x16 D spill (8 KB)

    const int lane = threadIdx.x & 31;
    const int wave = threadIdx.x >> 5;
    const int r16  = lane & 15;         // row within 16-row tile
    const int half = lane >> 4;         // lane-pair half
    const int myRow = blockIdx.x * 128 + wave * 16 + r16;
    const bool valid = (myRow < ROWS);

    const float* lrow = logits + (size_t)(valid ? myRow : 0) * NCLS;

    // row max (each row computed redundantly by its lane pair)
    float m = -1e30f;
    for (int k = 0; k < NCLS; ++k) m = fmaxf(m, lrow[k]);

    // stage exp values: half 0 -> k 0..41, half 1 -> k 42..83.
    // Loads are unconditional (clamped index); masking only on stored value.
    for (int k = half * 42; k < half * 42 + 42; ++k) {
        const int kc = (k < NCLS) ? k : (NCLS - 1);
        const float x = lrow[kc];
        float e = __expf(x - m);
        if (!valid || k >= NCLS) e = 0.0f;
        exl[wave][r16][k] = e;
    }
    __syncthreads();

    float rowsum;
#if USE_WMMA
    // A-matrix 16x4 f32 layout: lanes 0-15 hold A[M=lane][K=0..1] in 2 VGPRs,
    // lanes 16-31 hold A[M=lane-16][K=2..3].  B = all-ones (4x16) so D[i][j]
    // = rowsum(A_i) for every j.  21 chunks cover K=0..83 (pad contributes 0).
    v8f acc = (v8f){0.f,0.f,0.f,0.f,0.f,0.f,0.f,0.f};
    v2f ones; ones.x = 1.0f; ones.y = 1.0f;
#pragma unroll
    for (int c = 0; c < 21; ++c) {
        const int k0 = 4 * c + 2 * half;
        v2f a;
        a.x = exl[wave][r16][k0];
        a.y = exl[wave][r16][k0 + 1];
        acc = __builtin_amdgcn_wmma_f32_16x16x4_f32(
            /*neg_a=*/false, a, /*neg_b=*/false, ones,
            /*c_mod=*/(short)0, acc, /*reuse_a=*/false, /*reuse_b=*/false);
    }
    // C/D layout: VGPR r, lanes 0-15 -> (M=r, N=lane); lanes 16-31 -> (M=r+8).
#pragma unroll
    for (int r = 0; r < 8; ++r) tile[wave][r + 8 * half][r16] = acc[r];
    __syncthreads();
    rowsum = tile[wave][r16][0];
#else
    float s = 0.0f;
    for (int k = 0; k < KPAD; ++k) s += exl[wave][r16][k];
    rowsum = s;
    __syncthreads();
#endif

    if (valid) {
        const int b = myRow / NPROP;
        const int n = myRow - b * NPROP;
        const float4 p = ((const float4*)prop)[myRow];
        const float px = 0.5f * (p.x + p.z);
        const float py = 0.5f * (p.y + p.w);
        const float pw = p.z - p.x;
        const float ph = p.w - p.y;
        const float inv = 1.0f / rowsum;
        const float* rrow = breg + (size_t)myRow * (CCLS * 4);
        // lane pair splits the 80 classes: half 0 -> c 0..39, half 1 -> 40..79
        for (int c = half * 40; c < half * 40 + 40; ++c) {
            const float sc = exl[wave][r16][c + 1] * inv;   // skip background
            scores_ws[(b * CCLS + c) * NPROP + n] = sc;

            const float4 t = ((const float4*)rrow)[c];
            const float gx = t.x * pw + px;
            const float gy = t.y * ph + py;
            const float gw = pw * __expf(t.z);
            const float gh = ph * __expf(t.w);
            float4 o;
            o.x = gx - 0.5f * gw; o.y = gy - 0.5f * gh;
            o.z = gx + 0.5f * gw; o.w = gy + 0.5f * gh;
            ((float4*)boxes_ws)[(size_t)(b * CCLS + c) * NPROP + n] = o;
        }
    }
}

// =====================================================================
// Kernel 2: one block per (image, class).  Stage scores+boxes into LDS
// (async-to-LDS on CDNA5), stable O(N^2) rank sort, truncated NMS scan
// over the valid prefix (scores sorted desc => valid is a prefix), then
// cumsum cap at 100 and masked [box,score] output.
// =====================================================================
__global__ __launch_bounds__(256)
void k_nms(const float* __restrict__ scores_ws,
           const float* __restrict__ boxes_ws,
           float* __restrict__ out)
{
    __shared__ float  ss[NPROP];        // scores (raw, then sorted)
    __shared__ float4 sb[NPROP];        // boxes  (raw, then sorted)
    __shared__ int    supp[NPROP];
    __shared__ int    keep[NPROP];
    __shared__ int    Vsh;

    const int bc  = blockIdx.x;         // b*C + c
    const int tid = threadIdx.x;
    const float* gsc = scores_ws + (size_t)bc * NPROP;
    const float* gbx = boxes_ws  + (size_t)bc * NPROP * 4;

#if USE_ASYNC
    for (int i = tid; i < NPROP; i += 256) {
        __builtin_amdgcn_global_load_async_to_lds_b32(
            (as1_i32*)(gsc + i), (as3_i32*)(&ss[i]), 0, 0);
        __builtin_amdgcn_global_load_async_to_lds_b128(
            (as1_i128*)(gbx + 4 * i), (as3_i128*)(&sb[i]), 0, 0);
    }
    __builtin_amdgcn_s_wait_asynccnt(0);
#else
    for (int i = tid; i < NPROP; i += 256) {
        ss[i] = gsc[i];
        sb[i] = ((const float4*)gbx)[i];
    }
#endif
    for (int i = tid; i < NPROP; i += 256) { supp[i] = 0; keep[i] = 0; }
    if (tid == 0) Vsh = 0;
    __syncthreads();

    // ---- stable descending rank sort (matches argsort(-scores, stable)) ----
    float  myS[4]; float4 myB[4]; int myR[4]; int myI[4]; int nOwn = 0;
    for (int i = tid; i < NPROP; i += 256) {
        myI[nOwn] = i; myS[nOwn] = ss[i]; myB[nOwn] = sb[i]; myR[nOwn] = 0; ++nOwn;
    }
    for (int j = 0; j < NPROP; ++j) {
        const float sj = ss[j];
        for (int t = 0; t < nOwn; ++t)
            myR[t] += (sj > myS[t]) || ((sj == myS[t]) && (j < myI[t]));
    }
    __syncthreads();                       // everyone done reading raw arrays
    for (int t = 0; t < nOwn; ++t) {       // scatter into sorted position
        ss[myR[t]] = myS[t];
        sb[myR[t]] = myB[t];
    }
    __syncthreads();

    // ---- valid prefix length V = #(score > 0.05) (sorted desc) ----
    for (int t = 0; t < nOwn; ++t) {
        const int j = myI[t];
        if (ss[j] > SCORE_T && (j == NPROP - 1 || ss[j + 1] <= SCORE_T)) Vsh = j + 1;
    }
    __syncthreads();
    const int V = Vsh;

    // cache owned sorted boxes in registers
    float4 jb[4]; float ja[4]; int jsupp[4];
    for (int t = 0; t < nOwn; ++t) {
        jb[t] = sb[myI[t]];
        ja[t] = (jb[t].z - jb[t].x) * (jb[t].w - jb[t].y);
        jsupp[t] = 0;
    }
    __syncthreads();

    // ---- greedy NMS scan over valid prefix only ----
    for (int i = 0; i < V; ++i) {
        const int si = supp[i];
        if (tid == 0) keep[i] = si ? 0 : 1;
        if (!si) {
            const float4 bi = sb[i];
            const float  ai = (bi.z - bi.x) * (bi.w - bi.y);
            for (int t = 0; t < nOwn; ++t) {
                const int j = myI[t];
                if (j > i && j < V && !jsupp[t]) {
                    const float lx = fmaxf(bi.x, jb[t].x);
                    const float ly = fmaxf(bi.y, jb[t].y);
                    const float rx = fminf(bi.z, jb[t].z);
                    const float ry = fminf(bi.w, jb[t].w);
                    const float iw = fmaxf(rx - lx, 0.0f);
                    const float ih = fmaxf(ry - ly, 0.0f);
                    const float inter = iw * ih;
                    const float iou = inter / (ai + ja[t] - inter);
                    if (iou > NMS_T) { jsupp[t] = 1; supp[j] = 1; }
                }
            }
        }
        __syncthreads();
    }

    // ---- cap at DETCAP by rank, write [box*m, score*m] ----
    for (int t = 0; t < nOwn; ++t) {
        const int j = myI[t];
        float o0 = 0.f, o1 = 0.f, o2 = 0.f, o3 = 0.f, o4 = 0.f;
        if (j < V && keep[j]) {
            int rk = 0;
            for (int q = 0; q <= j; ++q) rk += keep[q];
            if (rk <= DETCAP) {
                const float4 bj = sb[j];
                o0 = bj.x; o1 = bj.y; o2 = bj.z; o3 = bj.w; o4 = ss[j];
            }
        }
        float* po = out + ((size_t)bc * NPROP + j) * 5;
        po[0] = o0; po[1] = o1; po[2] = o2; po[3] = o3; po[4] = o4;
    }
}

// =====================================================================
extern "C" void kernel_launch(void* const* d_in, const int* in_sizes, int n_in,
                              void* d_out, int out_size, void* d_ws, size_t ws_size,
                              hipStream_t stream)
{
    (void)in_sizes; (void)n_in; (void)out_size; (void)ws_size;
    const float* logits = (const float*)d_in[0];   // [2000, 81]
    const float* breg   = (const float*)d_in[1];   // [2000, 320]
    const float* prop   = (const float*)d_in[2];   // [2, 1000, 4]

    float* scores_ws = (float*)d_ws;                          // 160000 floats
    float* boxes_ws  = scores_ws + BATCH * CCLS * NPROP;      // 640000 floats

    k_scores_decode<<<16, 256, 0, stream>>>(logits, breg, prop, scores_ws, boxes_ws);
    k_nms<<<BATCH * CCLS, 256, 0, stream>>>(scores_ws, boxes_ws, (float*)d_out);
}